// GTNormalLoss_9758165696612
// MI455X (gfx1250) — compile-verified
//
#include <hip/hip_runtime.h>

// ---------------------------------------------------------------------------
// GTNormalLoss for MI455X (gfx1250, wave32).
// pred: (8,4096,3) f32, gt_normals: (8,4096,3) f32 -> scalar f32.
//
// - Batch point cloud (48KB) prefetched once per block into LDS with
//   gfx1250 async global->LDS loads (ASYNCcnt / s_wait_asynccnt path).
// - Pairwise inner products via V_WMMA_F32_16X16X4_F32 (K=3 padded to 4).
// - d = |xi|^2 + |xj|^2 - 2<xi,xj>; per-row top-10 in VGPRs via unrolled
//   compare-select network; wave-private LDS staging with NO block barriers
//   in the hot loop (same-wave DS ops are in-order on CDNA5).
// - Closed-form 3x3 symmetric eigen-solve -> PCA normal -> cosine loss.
// ---------------------------------------------------------------------------

typedef float v2f __attribute__((ext_vector_type(2)));
typedef float v8f __attribute__((ext_vector_type(8)));

constexpr int BATCH   = 8;
constexpr int NPTS    = 4096;
constexpr int KNN     = 10;
constexpr int NTILES  = NPTS / 16;      // 256 column tiles
constexpr int WPB     = 8;              // waves per block
constexpr int THREADS = 32 * WPB;       // 256

__global__ void gtnl_init(float* out) { out[0] = 0.0f; }

// Insert (v,vi) into ascending sorted list kd/ki of length KNN, dropping max.
// Fully unrolled, static indexing only -> lives entirely in VGPRs.
__device__ __forceinline__ void topk_insert(float (&kd)[KNN], int (&ki)[KNN],
                                            float v, int vi) {
    if (v < kd[KNN - 1]) {
#pragma unroll
        for (int t = 0; t < KNN; ++t) {
            bool sw = v < kd[t];
            float td = kd[t]; int ti = ki[t];
            kd[t] = sw ? v  : td;
            ki[t] = sw ? vi : ti;
            v  = sw ? td : v;
            vi = sw ? ti : vi;
        }
    }
}

__launch_bounds__(THREADS)
__global__ void gtnl_kernel(const float* __restrict__ pred,
                            const float* __restrict__ gtn,
                            float* __restrict__ out) {
    // Whole batch's point cloud, shared by the 8 waves of this block.
    __shared__ __align__(16) float Ppack[NPTS * 3];   // 48 KB
    __shared__ float Cn2[NPTS];                       // 16 KB |p|^2
    __shared__ float stage[WPB][16][16];              //  8 KB dist staging
    __shared__ float stashD[WPB][16][KNN];            //  5 KB half-merge
    __shared__ int   stashI[WPB][16][KNN];            //  5 KB

    const int tid  = threadIdx.x;
    const int lane = tid & 31;
    const int wave = tid >> 5;

    // Block = 8 consecutive row tiles of ONE batch (256 tiles/batch, 8/block).
    const int blockTile0 = blockIdx.x * WPB;          // 0..2047, step 8
    const int b  = blockTile0 >> 8;                   // batch index
    const int r0 = ((blockTile0 & 255) + wave) * 16;  // this wave's first row

    const float* __restrict__ Pg = pred + (size_t)b * NPTS * 3;

    // ---- async prefetch: 48 KB global -> LDS via gfx1250 async path --------
    {
        // generic LDS pointer: low 32 bits are the LDS byte offset
        unsigned lds  = (unsigned)(size_t)(&Ppack[0]) + (unsigned)tid * 16u;
        unsigned goff = (unsigned)tid * 16u;
        const unsigned long long gbase = (unsigned long long)(size_t)Pg;
        constexpr unsigned stride = THREADS * 16u;    // 4096 B per sweep
        constexpr int sweeps = (NPTS * 3 * 4) / (int)stride;  // 12
#pragma unroll
        for (int i = 0; i < sweeps; ++i) {
            asm volatile("global_load_async_to_lds_b128 %0, %1, %2"
                         :: "v"(lds), "v"(goff), "s"(gbase) : "memory");
            lds  += stride;
            goff += stride;
        }
        asm volatile("s_wait_asynccnt 0" ::: "memory");
    }
    __syncthreads();

    // ---- per-point squared norms (once per block) --------------------------
    for (int i = tid; i < NPTS; i += THREADS) {
        const float x = Ppack[i * 3 + 0];
        const float y = Ppack[i * 3 + 1];
        const float z = Ppack[i * 3 + 2];
        Cn2[i] = x * x + y * y + z * z;
    }
    __syncthreads();

    const int m    = lane & 15;
    const int half = lane >> 4;

    // ---- A operand: 16x4 f32 tile (rows r0..r0+15, coords x,y,z,0).
    // ISA layout: lanes 0-15 hold K=0,1 ; lanes 16-31 hold K=2,3 (same M).
    const float arx = Ppack[(r0 + m) * 3 + 0];
    const float ary = Ppack[(r0 + m) * 3 + 1];
    const float arz = Ppack[(r0 + m) * 3 + 2];
    v2f amat;
    amat.x = half ? arz : arx;
    amat.y = half ? 0.f : ary;

    // Squared norms of the 8 rows this lane's accumulator regs map to:
    // C VGPR r @ lane -> row r0 + r + 8*half.
    float rn2[8];
#pragma unroll
    for (int r = 0; r < 8; ++r)
        rn2[r] = Cn2[r0 + r + 8 * half];

    // Per-lane top-K (ascending). Lanes 0-15 scan cols 0-7 of each tile for
    // row m; lanes 16-31 scan cols 8-15 for the same row; merged afterwards.
    float kd[KNN]; int ki[KNN];
#pragma unroll
    for (int t = 0; t < KNN; ++t) { kd[t] = 3.4e38f; ki[t] = 0; }

    // ---- hot loop: WMMA distances + top-k. NO block barriers: the staging
    // buffer is wave-private and same-wave DS ops are in-order on CDNA5.
#pragma unroll 2
    for (int jt = 0; jt < NTILES; ++jt) {
        const int c0   = jt * 16;
        const int cidx = c0 + m;
        // B operand: 4x16 tile. VGPR0 lanes 0-15 = K0, 16-31 = K2; VGPR1 K1/K3.
        const float bcx = Ppack[cidx * 3 + 0];
        const float bcy = Ppack[cidx * 3 + 1];
        const float bcz = Ppack[cidx * 3 + 2];
        v2f bmat;
        bmat.x = half ? bcz : bcx;
        bmat.y = half ? 0.f : bcy;
        const float cn2 = Cn2[cidx];

        v8f acc = {};
        acc = __builtin_amdgcn_wmma_f32_16x16x4_f32(
            /*neg_a=*/false, amat, /*neg_b=*/false, bmat,
            /*c_mod=*/(short)0, acc, /*reuse_a=*/false, /*reuse_b=*/false);

        // acc[r] @ lane = <row r0+r+8*half , col c0+m>; stage so each lane
        // reads half of one row (row m) of the 16x16 distance tile.
#pragma unroll
        for (int r = 0; r < 8; ++r) {
            const float d = rn2[r] + cn2 - 2.0f * acc[r];
            stage[wave][r + 8 * half][m] = d;
        }
        asm volatile("" ::: "memory");   // keep DS store->load program order

        // Row m, candidate cols 8*half .. 8*half+7 of this tile.
#pragma unroll
        for (int cc = 0; cc < 8; ++cc) {
            const int col = 8 * half + cc;
            topk_insert(kd, ki, stage[wave][m][col], c0 + col);
        }
        asm volatile("" ::: "memory");   // reads done before next tile's stores
    }

    // ---- merge the two half-lane top-K lists (lane L / L+16 share row m) ---
    if (half) {
#pragma unroll
        for (int t = 0; t < KNN; ++t) {
            stashD[wave][m][t] = kd[t];
            stashI[wave][m][t] = ki[t];
        }
    }
    __syncthreads();

    float loss = 0.0f;
    if (!half) {
#pragma unroll
        for (int t = 0; t < KNN; ++t)
            topk_insert(kd, ki, stashD[wave][m][t], stashI[wave][m][t]);

        // ---- gather neighborhood from LDS, covariance (divide by KNN) ------
        float px[KNN], py[KNN], pz[KNN];
        float mx = 0.f, my = 0.f, mz = 0.f;
#pragma unroll
        for (int t = 0; t < KNN; ++t) {
            const int q = ki[t] * 3;
            px[t] = Ppack[q + 0]; py[t] = Ppack[q + 1]; pz[t] = Ppack[q + 2];
            mx += px[t]; my += py[t]; mz += pz[t];
        }
        const float invk = 1.0f / (float)KNN;
        mx *= invk; my *= invk; mz *= invk;

        float cxx = 0.f, cxy = 0.f, cxz = 0.f, cyy = 0.f, cyz = 0.f, czz = 0.f;
#pragma unroll
        for (int t = 0; t < KNN; ++t) {
            const float dx = px[t] - mx, dy = py[t] - my, dz = pz[t] - mz;
            cxx += dx * dx; cxy += dx * dy; cxz += dx * dz;
            cyy += dy * dy; cyz += dy * dz; czz += dz * dz;
        }
        cxx *= invk; cxy *= invk; cxz *= invk;
        cyy *= invk; cyz *= invk; czz *= invk;

        // ---- smallest eigenvalue of symmetric 3x3 (trigonometric form) -----
        const float q3  = (cxx + cyy + czz) * (1.0f / 3.0f);
        const float p1  = cxy * cxy + cxz * cxz + cyz * cyz;
        const float a00 = cxx - q3, a11 = cyy - q3, a22 = czz - q3;
        const float p2  = a00 * a00 + a11 * a11 + a22 * a22 + 2.0f * p1;
        const float pp  = sqrtf(p2 * (1.0f / 6.0f));
        float lmin;
        if (pp < 1e-20f) {
            lmin = q3;
        } else {
            const float ip = 1.0f / pp;
            const float b00 = a00 * ip, b01 = cxy * ip, b02 = cxz * ip;
            const float b11 = a11 * ip, b12 = cyz * ip, b22 = a22 * ip;
            float detB = b00 * (b11 * b22 - b12 * b12)
                       - b01 * (b01 * b22 - b12 * b02)
                       + b02 * (b01 * b12 - b11 * b02);
            float rr = fminf(1.0f, fmaxf(-1.0f, detB * 0.5f));
            const float phi = acosf(rr) * (1.0f / 3.0f);
            lmin = q3 + 2.0f * pp * cosf(phi + 2.0943951023931953f); // +2pi/3
        }

        // ---- eigenvector for lmin: best cross product of rows of (C-lmin I)
        const float r0x = cxx - lmin, r0y = cxy,        r0z = cxz;
        const float r1x = cxy,        r1y = cyy - lmin, r1z = cyz;
        const float r2x = cxz,        r2y = cyz,        r2z = czz - lmin;

        float c0x = r0y * r1z - r0z * r1y, c0y = r0z * r1x - r0x * r1z, c0z = r0x * r1y - r0y * r1x;
        float c1x = r0y * r2z - r0z * r2y, c1y = r0z * r2x - r0x * r2z, c1z = r0x * r2y - r0y * r2x;
        float c2x = r1y * r2z - r1z * r2y, c2y = r1z * r2x - r1x * r2z, c2z = r1x * r2y - r1y * r2x;
        const float n0 = c0x * c0x + c0y * c0y + c0z * c0z;
        const float n1 = c1x * c1x + c1y * c1y + c1z * c1z;
        const float n2 = c2x * c2x + c2y * c2y + c2z * c2z;

        float nx = c0x, ny = c0y, nz = c0z, best = n0;
        if (n1 > best) { best = n1; nx = c1x; ny = c1y; nz = c1z; }
        if (n2 > best) { best = n2; nx = c2x; ny = c2y; nz = c2z; }
        const float nn = sqrtf(best);
        if (nn > 1e-20f) { const float inn = 1.0f / nn; nx *= inn; ny *= inn; nz *= inn; }
        else             { nx = 0.f; ny = 0.f; nz = 1.f; }

        // ---- cosine loss vs gt normal (pred normal is unit; eps on gt norm)
        const float* g = gtn + ((size_t)b * NPTS + r0 + m) * 3;
        const float gx = g[0], gy = g[1], gz = g[2];
        const float dot = nx * gx + ny * gy + nz * gz;
        const float nb  = sqrtf(gx * gx + gy * gy + gz * gz);
        const float cosv = dot / (fmaxf(1.0f, 1e-8f) * fmaxf(nb, 1e-8f));
        loss = (1.0f - cosv) * (1.0f / (float)(BATCH * NPTS));
    }

    // ---- wave reduce (lanes 16-31 carry 0) + one atomic per wave -----------
#pragma unroll
    for (int off = 16; off > 0; off >>= 1)
        loss += __shfl_down(loss, off, 32);
    if (lane == 0) atomicAdd(out, loss);
}

extern "C" void kernel_launch(void* const* d_in, const int* in_sizes, int n_in,
                              void* d_out, int out_size, void* d_ws, size_t ws_size,
                              hipStream_t stream) {
    const float* pred = (const float*)d_in[0];
    const float* gtn  = (const float*)d_in[1];
    float* out = (float*)d_out;

    gtnl_init<<<dim3(1), dim3(1), 0, stream>>>(out);

    const int totalWaves = BATCH * (NPTS / 16);           // 2048
    const int blocks     = totalWaves / WPB;              // 256
    gtnl_kernel<<<dim3(blocks), dim3(THREADS), 0, stream>>>(pred, gtn, out);
}